// MultiHeadDotAttLayer_38714835206282
// MI455X (gfx1250) — compile-verified
//
#include <hip/hip_runtime.h>
#include <hip/hip_bf16.h>

// ---------------------------------------------------------------------------
// Multi-head sparse graph attention for MI455X (gfx1250, wave32).
//   h:        [N, 128]  f32
//   edge_idx: [2, E]    int32 (row = dst, col = src)
//   WQ/WK/WV: [8, 128, 32] f32
//   out:      [N, 256]  f32   (head-major concat)
//
// Design: Q/K/V stored node-major [N][256] (153.6 MB -> fits 192 MB L2, so
// per-edge 1KB gathers are L2-resident). Projections use V_WMMA_F32_16X16X4_F32
// (exact f32 math) with the h tile staged via the gfx1250 async global->LDS
// path (ASYNCcnt). Segment softmax via atomics.
// ---------------------------------------------------------------------------

typedef __attribute__((ext_vector_type(2))) float v2f;
typedef __attribute__((ext_vector_type(8))) float v8f;

#define DIN   128
#define DOUT  256   // 8 heads * 32
#define NHEAD 8

// ---------------------------------------------------------------------------
// Init: out = 0, denom = 0, m = -inf
// ---------------------------------------------------------------------------
__global__ void init_buffers(float* __restrict__ out, float* __restrict__ m,
                             float* __restrict__ denom, long long outN, long long nh)
{
    long long i = (long long)blockIdx.x * blockDim.x + threadIdx.x;
    if (i < outN) out[i] = 0.0f;
    if (i < nh) { m[i] = -__builtin_inff(); denom[i] = 0.0f; }
}

// ---------------------------------------------------------------------------
// Projection GEMM:  P[n, h*32+k] = sum_d h[n,d] * W[h,d,k]  for Q, K, V.
// Block = 512 threads (16 wave32). Block covers 16 rows x 256 cols.
// Each wave owns a 16x16 tile; K-loop in steps of 4 using f32 WMMA.
//
// Fragment layouts (wave32, 16x16x4 f32 per ISA 7.12.2):
//   A (16x4): lane L -> row M = L&15, K pair = (L>>4)*2; vgpr0=K, vgpr1=K+1
//   B (4x16): lane L -> col N = L&15, K pair = (L>>4)*2; vgpr0=K, vgpr1=K+1
//   C/D     : lane L, vgpr r -> row M = r + 8*(L>>4), col N = L&15
// ---------------------------------------------------------------------------
__global__ __launch_bounds__(512)
void proj_qkv(const float* __restrict__ h,
              const float* __restrict__ WQ,
              const float* __restrict__ WK,
              const float* __restrict__ WV,
              float* __restrict__ Q,
              float* __restrict__ Kp,
              float* __restrict__ Vp)
{
    __shared__ float hs[16 * 132];          // 16x128 tile, padded to kill bank conflicts

    const int tid  = threadIdx.x;
    const int row0 = blockIdx.x * 16;

    // Async global->LDS copy of the 16x128 h tile: each of the 512 threads
    // moves one 16-byte chunk directly into LDS (no VGPR round trip).
    // The VDST operand carries the wave-relative LDS byte offset; we derive it
    // from the element's generic pointer (low 32 bits of a flat LDS pointer ==
    // group-relative offset on amdgcn). The ptrtoint also CAPTURES hs, so the
    // "memory" clobber forces the compiler to reload hs after the copy instead
    // of folding the (otherwise provably-unwritten) LDS reads to undef.
    {
        const int r  = tid >> 5;                    // 0..15 tile row
        const int c4 = (tid & 31) * 4;              // float column 0,4,...,124
        const unsigned long long gaddr =
            (unsigned long long)(const void*)(h + (size_t)(row0 + r) * DIN + c4);
        const unsigned lds_off =
            (unsigned)(unsigned long long)(const void*)(&hs[r * 132 + c4]);
        asm volatile("global_load_async_to_lds_b128 %0, %1, off"
                     :: "v"(lds_off), "v"(gaddr) : "memory");
        asm volatile("s_wait_asynccnt 0x0" ::: "memory");
    }
    __syncthreads();

    const int wave = tid >> 5;              // 0..15 -> 16-col tile
    const int lane = tid & 31;
    const int n    = lane & 15;             // tile row (A) / tile col (B, D)
    const int kg   = lane >> 4;             // selects K pair {0,1} vs {2,3}
    const int col0 = wave * 16;
    const int head = col0 >> 5;             // all 16 cols of a wave share one head
    const int kin  = (col0 & 31) + n;       // column within the head's 32

    const float* bq = WQ + (size_t)head * (DIN * 32) + kin;
    const float* bk = WK + (size_t)head * (DIN * 32) + kin;
    const float* bv = WV + (size_t)head * (DIN * 32) + kin;

    v8f cq = {}; v8f ck = {}; v8f cv = {};

    for (int k0 = 0; k0 < DIN; k0 += 4) {
        const int ka = k0 + kg * 2;
        v2f a;  a.x  = hs[n * 132 + ka];        a.y  = hs[n * 132 + ka + 1];
        v2f bQ; bQ.x = bq[(size_t)ka * 32];     bQ.y = bq[(size_t)(ka + 1) * 32];
        v2f bK; bK.x = bk[(size_t)ka * 32];     bK.y = bk[(size_t)(ka + 1) * 32];
        v2f bV; bV.x = bv[(size_t)ka * 32];     bV.y = bv[(size_t)(ka + 1) * 32];

        cq = __builtin_amdgcn_wmma_f32_16x16x4_f32(false, a, false, bQ, (short)0, cq, false, false);
        ck = __builtin_amdgcn_wmma_f32_16x16x4_f32(false, a, false, bK, (short)0, ck, false, false);
        cv = __builtin_amdgcn_wmma_f32_16x16x4_f32(false, a, false, bV, (short)0, cv, false, false);
    }

    const int colg = col0 + n;
    #pragma unroll
    for (int r = 0; r < 8; ++r) {
        const size_t o = (size_t)(row0 + r + 8 * kg) * DOUT + colg;
        Q[o]  = cq[r];
        Kp[o] = ck[r];
        Vp[o] = cv[r];
    }
}

// ---------------------------------------------------------------------------
// Float atomic max via sign-aware integer atomics (works with -inf init).
// ---------------------------------------------------------------------------
__device__ inline void atomicMaxFloat(float* addr, float val)
{
    if (val >= 0.0f) atomicMax((int*)addr, __float_as_int(val));
    else             atomicMin((unsigned int*)addr, __float_as_uint(val));
}

// ---------------------------------------------------------------------------
// Pass C: per-edge logits for all 8 heads + segment max.  One wave per edge.
// Lane l holds Q/K elements [8l, 8l+8); head hd lives in lanes 4*hd..4*hd+3.
// ---------------------------------------------------------------------------
__global__ __launch_bounds__(256)
void edge_logits_max(const float* __restrict__ Q, const float* __restrict__ Kp,
                     const int* __restrict__ row_idx, const int* __restrict__ col_idx,
                     float* __restrict__ logits, float* __restrict__ m, long long E)
{
    const long long e = (long long)blockIdx.x * 8 + (threadIdx.x >> 5);
    if (e >= E) return;
    const int lane = threadIdx.x & 31;
    const int r = row_idx[e];
    const int c = col_idx[e];

    const float4* Qr = (const float4*)(Q  + (size_t)r * DOUT);
    const float4* Kc = (const float4*)(Kp + (size_t)c * DOUT);
    float4 qa = Qr[lane * 2], qb = Qr[lane * 2 + 1];
    float4 ka = Kc[lane * 2], kb = Kc[lane * 2 + 1];

    float s = qa.x * ka.x + qa.y * ka.y + qa.z * ka.z + qa.w * ka.w
            + qb.x * kb.x + qb.y * kb.y + qb.z * kb.z + qb.w * kb.w;
    s += __shfl_xor(s, 1, 32);
    s += __shfl_xor(s, 2, 32);

    if ((lane & 3) == 0) {
        const int hd = lane >> 2;
        const float lg = s * 0.17677669529663689f;   // 1/sqrt(32)
        logits[e * NHEAD + hd] = lg;
        atomicMaxFloat(&m[(size_t)r * NHEAD + hd], lg);
    }
}

// ---------------------------------------------------------------------------
// Pass D: ex = exp(logit - m[row]); denom[row] += ex; logits <- ex (in place).
// One thread per (edge, head).
// ---------------------------------------------------------------------------
__global__ __launch_bounds__(256)
void edge_exp_denom(const int* __restrict__ row_idx, float* __restrict__ logits,
                    const float* __restrict__ m, float* __restrict__ denom, long long EH)
{
    const long long i = (long long)blockIdx.x * blockDim.x + threadIdx.x;
    if (i >= EH) return;
    const long long e = i >> 3;
    const int hd = (int)(i & 7);
    const int r = row_idx[e];
    const float ex = expf(logits[i] - m[(size_t)r * NHEAD + hd]);
    logits[i] = ex;
    atomicAdd(&denom[(size_t)r * NHEAD + hd], ex);
}

// ---------------------------------------------------------------------------
// Pass E: out[row] += (ex/denom[row]) * V[col].  One wave per edge; lane l
// owns output floats [8l, 8l+8) (all within head l>>2).
// ---------------------------------------------------------------------------
__global__ __launch_bounds__(256)
void edge_aggregate(const float* __restrict__ Vp,
                    const int* __restrict__ row_idx, const int* __restrict__ col_idx,
                    const float* __restrict__ ex, const float* __restrict__ denom,
                    float* __restrict__ out, long long E)
{
    const long long e = (long long)blockIdx.x * 8 + (threadIdx.x >> 5);
    if (e >= E) return;
    const int lane = threadIdx.x & 31;
    const int r = row_idx[e];
    const int c = col_idx[e];
    const int hd = lane >> 2;

    const float att = ex[e * NHEAD + hd] / denom[(size_t)r * NHEAD + hd];

    const float4* Vc = (const float4*)(Vp + (size_t)c * DOUT);
    float4 va = Vc[lane * 2], vb = Vc[lane * 2 + 1];

    float* o = out + (size_t)r * DOUT + lane * 8;
    atomicAdd(o + 0, att * va.x);
    atomicAdd(o + 1, att * va.y);
    atomicAdd(o + 2, att * va.z);
    atomicAdd(o + 3, att * va.w);
    atomicAdd(o + 4, att * vb.x);
    atomicAdd(o + 5, att * vb.y);
    atomicAdd(o + 6, att * vb.z);
    atomicAdd(o + 7, att * vb.w);
}

// ---------------------------------------------------------------------------
extern "C" void kernel_launch(void* const* d_in, const int* in_sizes, int n_in,
                              void* d_out, int out_size, void* d_ws, size_t ws_size,
                              hipStream_t stream)
{
    const float* h    = (const float*)d_in[0];
    const int*   eidx = (const int*)d_in[1];     // [2, E] int32
    const float* WQ   = (const float*)d_in[2];
    const float* WK   = (const float*)d_in[3];
    const float* WV   = (const float*)d_in[4];

    const long long N = in_sizes[0] / DIN;       // 50000
    const long long E = in_sizes[1] / 2;         // 1,600,000
    const int* row = eidx;
    const int* col = eidx + E;

    // Workspace carve-up (floats): Q,K,V [N*256] each; m,denom [N*8]; ex [E*8].
    float* ws    = (float*)d_ws;
    const size_t nf = (size_t)N * DOUT;
    float* Q     = ws;
    float* Kp    = Q + nf;
    float* Vp    = Kp + nf;
    float* m     = Vp + nf;
    float* denom = m + (size_t)N * NHEAD;
    float* ex    = denom + (size_t)N * NHEAD;
    float* out   = (float*)d_out;

    // 1) init out / m / denom
    {
        const long long outN = (long long)out_size;
        const long long nh   = N * NHEAD;
        const long long mx   = (outN > nh) ? outN : nh;
        init_buffers<<<(unsigned)((mx + 255) / 256), 256, 0, stream>>>(out, m, denom, outN, nh);
    }

    // 2) Q/K/V projections via f32 WMMA (N = 50000 is a multiple of 16)
    proj_qkv<<<(unsigned)(N / 16), 512, 0, stream>>>(h, WQ, WK, WV, Q, Kp, Vp);

    // 3) edge logits + segment max
    edge_logits_max<<<(unsigned)((E + 7) / 8), 256, 0, stream>>>(Q, Kp, row, col, ex, m, E);

    // 4) exp + segment sum
    edge_exp_denom<<<(unsigned)((E * NHEAD + 255) / 256), 256, 0, stream>>>(row, ex, m, denom, E * NHEAD);

    // 5) normalize + weighted scatter into out
    edge_aggregate<<<(unsigned)((E + 7) / 8), 256, 0, stream>>>(Vp, row, col, ex, denom, out, E);
}